// DeepseekV3MoE_17806934409994
// MI455X (gfx1250) — compile-verified
//
#include <hip/hip_runtime.h>
#include <hip/hip_bf16.h>
#include <cstddef>

// ---------------------------------------------------------------------------
// DeepseekV3 MoE for gfx1250 (MI455X), fp32 via V_WMMA_F32_16X16X4_F32.
// Sparse top-4 routing (4096 assignments) instead of the reference dense scan.
// GEMMs: 32x64 block tiles, K-chunk 32, double-buffered LDS fed by
// GLOBAL_LOAD_ASYNC_TO_LDS_B128 (ASYNCcnt) when the toolchain exposes it.
// ---------------------------------------------------------------------------

typedef __attribute__((ext_vector_type(2))) float v2f;
typedef __attribute__((ext_vector_type(8))) float v8f;
typedef __attribute__((__vector_size__(4 * sizeof(int)))) int v4i;

#define T_TOK   1024
#define D_DIM   2048
#define E_EXP   32
#define I_DIM   1408
#define SHI_DIM 2816
#define TOPK    4
#define NASSIGN (T_TOK * TOPK)

#if __has_builtin(__builtin_amdgcn_global_load_async_to_lds_b128) && \
    __has_builtin(__builtin_amdgcn_s_wait_asynccnt)
#define USE_ASYNC 1
#else
#define USE_ASYNC 0
#endif

__device__ __forceinline__ void cp_b128(const float* g, float* l) {
#if USE_ASYNC
  __builtin_amdgcn_global_load_async_to_lds_b128(
      (__attribute__((address_space(1))) v4i*)g,
      (__attribute__((address_space(3))) v4i*)l, 0, 0);
#else
  *(float4*)l = *(const float4*)g;
#endif
}

template <int N>
__device__ __forceinline__ void wait_stage() {
#if USE_ASYNC
  __builtin_amdgcn_s_wait_asynccnt(N);
#endif
}

__device__ __forceinline__ v8f wmma_f32(v2f a, v2f b, v8f c) {
  // (neg_a, A, neg_b, B, c_mod, C, reuse_a, reuse_b)
  return __builtin_amdgcn_wmma_f32_16x16x4_f32(false, a, false, b, (short)0, c,
                                               false, false);
}

__device__ __forceinline__ float silu_f(float x) {
  return x / (1.0f + __expf(-x));
}

// ---------------------------------------------------------------------------
// Kernel 1: gate + biased grouped top-k routing. One wave (32 lanes) per token.
// ---------------------------------------------------------------------------
__global__ __launch_bounds__(32) void gate_route_kernel(
    const float* __restrict__ x, const float* __restrict__ gate_w,
    const float* __restrict__ gate_b, int* __restrict__ counts,
    int* __restrict__ tok_list, float* __restrict__ wt_list,
    int* __restrict__ tk_e, int* __restrict__ tk_s) {
  const int t = blockIdx.x;
  const int lane = threadIdx.x;  // lane == expert index (E==32==wave32)

  const float* xr = x + (size_t)t * D_DIM;
  const float* wr = gate_w + (size_t)lane * D_DIM;
  float acc = 0.f;
  for (int d = 0; d < D_DIM; d += 4) {
    float4 xv = *(const float4*)(xr + d);
    float4 wv = *(const float4*)(wr + d);
    acc += xv.x * wv.x + xv.y * wv.y + xv.z * wv.z + xv.w * wv.w;
  }
  const float b = gate_b[lane];
  const float score = 1.f / (1.f + __expf(-(acc + b)));  // sigmoid(logit)
  const float biased = score + b;                        // score + bias

  __shared__ float sc[E_EXP];
  __shared__ float bi[E_EXP];
  sc[lane] = score;
  bi[lane] = biased;
  __syncthreads();

  if (lane == 0) {
    // group score = sum of top-2 biased scores within each group of 4
    float gs[8];
    for (int g = 0; g < 8; ++g) {
      float v[4] = {bi[4 * g + 0], bi[4 * g + 1], bi[4 * g + 2], bi[4 * g + 3]};
      int im = 0;
      float m1 = v[0];
      for (int i = 1; i < 4; ++i)
        if (v[i] > m1) { m1 = v[i]; im = i; }
      float m2 = -1e30f;
      for (int i = 0; i < 4; ++i)
        if (i != im && v[i] > m2) m2 = v[i];
      gs[g] = m1 + m2;
    }
    // top-4 groups (strict >, first index wins: matches lax.top_k)
    bool gsel[8] = {false, false, false, false, false, false, false, false};
    for (int k = 0; k < 4; ++k) {
      int bg = 0;
      float best = -1e30f;
      for (int g = 0; g < 8; ++g)
        if (!gsel[g] && gs[g] > best) { best = gs[g]; bg = g; }
      gsel[bg] = true;
    }
    // mask + top-4 experts
    float tmp[E_EXP];
    for (int e = 0; e < E_EXP; ++e) tmp[e] = gsel[e >> 2] ? bi[e] : 0.0f;
    int ids[TOPK];
    for (int k = 0; k < TOPK; ++k) {
      int bidx = 0;
      float best = -1e30f;
      for (int e = 0; e < E_EXP; ++e)
        if (tmp[e] > best) { best = tmp[e]; bidx = e; }
      ids[k] = bidx;
      tmp[bidx] = -1e30f;
    }
    float sw = 0.f;
    for (int k = 0; k < TOPK; ++k) sw += sc[ids[k]];
    const float inv = 2.5f / sw;  // fold ROUTED_SCALING into the weight
    for (int k = 0; k < TOPK; ++k) {
      const int e = ids[k];
      const int slot = atomicAdd(&counts[e], 1);
      tok_list[e * T_TOK + slot] = t;
      wt_list[e * T_TOK + slot] = sc[e] * inv;
      tk_e[t * TOPK + k] = e;
      tk_s[t * TOPK + k] = slot;
    }
  }
}

// ---------------------------------------------------------------------------
// Kernel 2: exclusive scan of counts (E=32, trivial).
// ---------------------------------------------------------------------------
__global__ void scan_kernel(const int* __restrict__ counts,
                            int* __restrict__ offsets) {
  if (threadIdx.x == 0 && blockIdx.x == 0) {
    int s = 0;
    for (int e = 0; e < E_EXP; ++e) {
      offsets[e] = s;
      s += counts[e];
    }
    offsets[E_EXP] = s;
  }
}

// ---------------------------------------------------------------------------
// Shared expert fused gate/up GEMM: Hs = silu(x@sw_gate) * (x@sw_up)
// grid = (SHI/64, T/32); 5 async b128 per thread per stage.
// ---------------------------------------------------------------------------
__global__ __launch_bounds__(256) void shared_up_kernel(
    const float* __restrict__ X, const float* __restrict__ Wg,
    const float* __restrict__ Wu, float* __restrict__ Hs) {
  const int mt = blockIdx.y, nt = blockIdx.x;
  const int nbase = nt * 64;
  __shared__ float As[2][32][36];
  __shared__ float Bg[2][32][68];
  __shared__ float Bu[2][32][68];
  const int tid = threadIdx.x;
  const int lane = tid & 31, wid = tid >> 5;
  const int wm = wid >> 2, wn = wid & 3;
  const int l = lane & 15, hi = lane >> 4;
  const int ar = tid >> 3, ac = (tid & 7) * 4;
  const int br = tid >> 4, bc = (tid & 15) * 4;
  const float* arow = X + (size_t)(mt * 32 + ar) * D_DIM + ac;

  auto stage = [&](int kc, int buf) {
    cp_b128(arow + kc, &As[buf][ar][ac]);
#pragma unroll
    for (int j = 0; j < 2; ++j) {
      const int rr = br + 16 * j;
      cp_b128(Wg + (size_t)(kc + rr) * SHI_DIM + nbase + bc, &Bg[buf][rr][bc]);
      cp_b128(Wu + (size_t)(kc + rr) * SHI_DIM + nbase + bc, &Bu[buf][rr][bc]);
    }
  };

  v8f cg = {};
  v8f cu = {};
  const int nch = D_DIM / 32;
  stage(0, 0);
  for (int i = 0; i < nch; ++i) {
    const int buf = i & 1;
    if (i + 1 < nch) {
      stage((i + 1) * 32, buf ^ 1);
      wait_stage<5>();
    } else {
      wait_stage<0>();
    }
    __syncthreads();
#pragma unroll
    for (int kk = 0; kk < 32; kk += 4) {
      const int k0 = kk + 2 * hi;
      v2f a;  a.x = As[buf][16 * wm + l][k0];  a.y = As[buf][16 * wm + l][k0 + 1];
      v2f bg; bg.x = Bg[buf][k0][16 * wn + l]; bg.y = Bg[buf][k0 + 1][16 * wn + l];
      v2f bu; bu.x = Bu[buf][k0][16 * wn + l]; bu.y = Bu[buf][k0 + 1][16 * wn + l];
      cg = wmma_f32(a, bg, cg);
      cu = wmma_f32(a, bu, cu);
    }
    __syncthreads();
  }
#pragma unroll
  for (int v = 0; v < 8; ++v) {
    const int row = mt * 32 + 16 * wm + v + 8 * hi;
    const int col = nbase + 16 * wn + l;
    Hs[(size_t)row * SHI_DIM + col] = silu_f(cg[v]) * cu[v];
  }
}

// ---------------------------------------------------------------------------
// Shared expert down GEMM: out = Hs @ sw_down (plain store; routed adds later)
// grid = (D/64, T/32); 3 async b128 per thread per stage.
// ---------------------------------------------------------------------------
__global__ __launch_bounds__(256) void shared_down_kernel(
    const float* __restrict__ Hs, const float* __restrict__ Wd,
    float* __restrict__ out) {
  const int mt = blockIdx.y, nt = blockIdx.x;
  const int nbase = nt * 64;
  __shared__ float As[2][32][36];
  __shared__ float Bd[2][32][68];
  const int tid = threadIdx.x;
  const int lane = tid & 31, wid = tid >> 5;
  const int wm = wid >> 2, wn = wid & 3;
  const int l = lane & 15, hi = lane >> 4;
  const int ar = tid >> 3, ac = (tid & 7) * 4;
  const int br = tid >> 4, bc = (tid & 15) * 4;
  const float* arow = Hs + (size_t)(mt * 32 + ar) * SHI_DIM + ac;

  auto stage = [&](int kc, int buf) {
    cp_b128(arow + kc, &As[buf][ar][ac]);
#pragma unroll
    for (int j = 0; j < 2; ++j) {
      const int rr = br + 16 * j;
      cp_b128(Wd + (size_t)(kc + rr) * D_DIM + nbase + bc, &Bd[buf][rr][bc]);
    }
  };

  v8f c = {};
  const int nch = SHI_DIM / 32;
  stage(0, 0);
  for (int i = 0; i < nch; ++i) {
    const int buf = i & 1;
    if (i + 1 < nch) {
      stage((i + 1) * 32, buf ^ 1);
      wait_stage<3>();
    } else {
      wait_stage<0>();
    }
    __syncthreads();
#pragma unroll
    for (int kk = 0; kk < 32; kk += 4) {
      const int k0 = kk + 2 * hi;
      v2f a;  a.x = As[buf][16 * wm + l][k0];  a.y = As[buf][16 * wm + l][k0 + 1];
      v2f bd; bd.x = Bd[buf][k0][16 * wn + l]; bd.y = Bd[buf][k0 + 1][16 * wn + l];
      c = wmma_f32(a, bd, c);
    }
    __syncthreads();
  }
#pragma unroll
  for (int v = 0; v < 8; ++v) {
    const int row = mt * 32 + 16 * wm + v + 8 * hi;
    const int col = nbase + 16 * wn + l;
    out[(size_t)row * D_DIM + col] = c[v];
  }
}

// ---------------------------------------------------------------------------
// Routed fused gate/up GEMM on gathered token rows:
//   H[offsets[e]+slot] = silu(x_tok @ w_gate[e]) * (x_tok @ w_up[e])
// grid = (I/64, T/32 worst-case m-tiles, E); early-exit past count[e].
// Out-of-range tile rows gather a clamped (valid) row: row m of A only feeds
// row m of C, and those C rows are never stored.
// ---------------------------------------------------------------------------
__global__ __launch_bounds__(256) void moe_up_kernel(
    const float* __restrict__ X, const float* __restrict__ Wgate,
    const float* __restrict__ Wup, const int* __restrict__ counts,
    const int* __restrict__ offsets, const int* __restrict__ tok_list,
    float* __restrict__ H) {
  const int e = blockIdx.z, mt = blockIdx.y, nt = blockIdx.x;
  const int count = counts[e];
  if (mt * 32 >= count) return;
  const int base = offsets[e];
  const float* wg = Wgate + (size_t)e * D_DIM * I_DIM;
  const float* wu = Wup + (size_t)e * D_DIM * I_DIM;
  const int nbase = nt * 64;

  __shared__ float As[2][32][36];
  __shared__ float Bg[2][32][68];
  __shared__ float Bu[2][32][68];
  const int tid = threadIdx.x;
  const int lane = tid & 31, wid = tid >> 5;
  const int wm = wid >> 2, wn = wid & 3;
  const int l = lane & 15, hi = lane >> 4;
  const int ar = tid >> 3, ac = (tid & 7) * 4;
  const int br = tid >> 4, bc = (tid & 15) * 4;

  int slot_a = mt * 32 + ar;
  if (slot_a > count - 1) slot_a = count - 1;  // clamp: result row discarded
  const float* arow =
      X + (size_t)tok_list[e * T_TOK + slot_a] * D_DIM + ac;

  auto stage = [&](int kc, int buf) {
    cp_b128(arow + kc, &As[buf][ar][ac]);
#pragma unroll
    for (int j = 0; j < 2; ++j) {
      const int rr = br + 16 * j;
      cp_b128(wg + (size_t)(kc + rr) * I_DIM + nbase + bc, &Bg[buf][rr][bc]);
      cp_b128(wu + (size_t)(kc + rr) * I_DIM + nbase + bc, &Bu[buf][rr][bc]);
    }
  };

  v8f cg = {};
  v8f cu = {};
  const int nch = D_DIM / 32;
  stage(0, 0);
  for (int i = 0; i < nch; ++i) {
    const int buf = i & 1;
    if (i + 1 < nch) {
      stage((i + 1) * 32, buf ^ 1);
      wait_stage<5>();
    } else {
      wait_stage<0>();
    }
    __syncthreads();
#pragma unroll
    for (int kk = 0; kk < 32; kk += 4) {
      const int k0 = kk + 2 * hi;
      v2f a;  a.x = As[buf][16 * wm + l][k0];  a.y = As[buf][16 * wm + l][k0 + 1];
      v2f bg; bg.x = Bg[buf][k0][16 * wn + l]; bg.y = Bg[buf][k0 + 1][16 * wn + l];
      v2f bu; bu.x = Bu[buf][k0][16 * wn + l]; bu.y = Bu[buf][k0 + 1][16 * wn + l];
      cg = wmma_f32(a, bg, cg);
      cu = wmma_f32(a, bu, cu);
    }
    __syncthreads();
  }
#pragma unroll
  for (int v = 0; v < 8; ++v) {
    const int mrow = 16 * wm + v + 8 * hi;
    const int slot = mt * 32 + mrow;
    if (slot < count) {
      const int col = nbase + 16 * wn + l;
      H[(size_t)(base + slot) * I_DIM + col] = silu_f(cg[v]) * cu[v];
    }
  }
}

// ---------------------------------------------------------------------------
// Routed down GEMM with per-row route weight (incl. 2.5):
//   R[offsets[e]+slot] = w * (H_row @ w_down[e])      (plain, deterministic)
// grid = (D/64, T/32 worst-case m-tiles, E)
// ---------------------------------------------------------------------------
__global__ __launch_bounds__(256) void moe_down_kernel(
    const float* __restrict__ H, const float* __restrict__ Wdown,
    const int* __restrict__ counts, const int* __restrict__ offsets,
    const float* __restrict__ wt_list, float* __restrict__ R) {
  const int e = blockIdx.z, mt = blockIdx.y, nt = blockIdx.x;
  const int count = counts[e];
  if (mt * 32 >= count) return;
  const int base = offsets[e];
  const float* wd = Wdown + (size_t)e * I_DIM * D_DIM;
  const int nbase = nt * 64;

  __shared__ float As[2][32][36];
  __shared__ float Bd[2][32][68];
  __shared__ float wrow[32];
  const int tid = threadIdx.x;
  if (tid < 32) {
    const int slot = mt * 32 + tid;
    wrow[tid] = (slot < count) ? wt_list[e * T_TOK + slot] : 0.f;
  }

  const int lane = tid & 31, wid = tid >> 5;
  const int wm = wid >> 2, wn = wid & 3;
  const int l = lane & 15, hi = lane >> 4;
  const int ar = tid >> 3, ac = (tid & 7) * 4;
  const int br = tid >> 4, bc = (tid & 15) * 4;

  int slot_a = mt * 32 + ar;
  if (slot_a > count - 1) slot_a = count - 1;  // clamp: result row discarded
  const float* arow = H + (size_t)(base + slot_a) * I_DIM + ac;

  auto stage = [&](int kc, int buf) {
    cp_b128(arow + kc, &As[buf][ar][ac]);
#pragma unroll
    for (int j = 0; j < 2; ++j) {
      const int rr = br + 16 * j;
      cp_b128(wd + (size_t)(kc + rr) * D_DIM + nbase + bc, &Bd[buf][rr][bc]);
    }
  };

  v8f c = {};
  const int nch = I_DIM / 32;
  stage(0, 0);
  for (int i = 0; i < nch; ++i) {
    const int buf = i & 1;
    if (i + 1 < nch) {
      stage((i + 1) * 32, buf ^ 1);
      wait_stage<3>();
    } else {
      wait_stage<0>();
    }
    __syncthreads();
#pragma unroll
    for (int kk = 0; kk < 32; kk += 4) {
      const int k0 = kk + 2 * hi;
      v2f a;  a.x = As[buf][16 * wm + l][k0];  a.y = As[buf][16 * wm + l][k0 + 1];
      v2f bd; bd.x = Bd[buf][k0][16 * wn + l]; bd.y = Bd[buf][k0 + 1][16 * wn + l];
      c = wmma_f32(a, bd, c);
    }
    __syncthreads();
  }
#pragma unroll
  for (int v = 0; v < 8; ++v) {
    const int mrow = 16 * wm + v + 8 * hi;
    const int slot = mt * 32 + mrow;
    if (slot < count) {
      const int col = nbase + 16 * wn + l;
      R[(size_t)(base + slot) * D_DIM + col] = c[v] * wrow[mrow];
    }
  }
}

// ---------------------------------------------------------------------------
// Deterministic combine: out[t] += sum_k R[offsets[e_k] + slot_k]
// ---------------------------------------------------------------------------
__global__ __launch_bounds__(256) void combine_kernel(
    const int* __restrict__ offsets, const int* __restrict__ tk_e,
    const int* __restrict__ tk_s, const float* __restrict__ R,
    float* __restrict__ out) {
  const int t = blockIdx.x;
  int rowi[TOPK];
#pragma unroll
  for (int k = 0; k < TOPK; ++k)
    rowi[k] = offsets[tk_e[t * TOPK + k]] + tk_s[t * TOPK + k];
  for (int d = threadIdx.x; d < D_DIM; d += 256) {
    float acc = out[(size_t)t * D_DIM + d];
#pragma unroll
    for (int k = 0; k < TOPK; ++k) acc += R[(size_t)rowi[k] * D_DIM + d];
    out[(size_t)t * D_DIM + d] = acc;
  }
}

// ---------------------------------------------------------------------------
// Host launcher
// ---------------------------------------------------------------------------
extern "C" void kernel_launch(void* const* d_in, const int* in_sizes, int n_in,
                              void* d_out, int out_size, void* d_ws,
                              size_t ws_size, hipStream_t stream) {
  const float* x       = (const float*)d_in[0];
  const float* gate_w  = (const float*)d_in[1];
  const float* gate_b  = (const float*)d_in[2];
  const float* w_gate  = (const float*)d_in[3];
  const float* w_up    = (const float*)d_in[4];
  const float* w_down  = (const float*)d_in[5];
  const float* sw_gate = (const float*)d_in[6];
  const float* sw_up   = (const float*)d_in[7];
  const float* sw_down = (const float*)d_in[8];
  float* out = (float*)d_out;

  // Workspace layout (all 256B-aligned); total ~66 MB.
  char* ws = (char*)d_ws;
  size_t off = 0;
  int* counts = (int*)(ws + off);   off += 256;
  int* offsets = (int*)(ws + off);  off += 256;
  int* tok_list = (int*)(ws + off); off += (size_t)E_EXP * T_TOK * 4;   // 128 KB
  float* wt_list = (float*)(ws + off); off += (size_t)E_EXP * T_TOK * 4;
  int* tk_e = (int*)(ws + off);     off += (size_t)NASSIGN * 4;
  int* tk_s = (int*)(ws + off);     off += (size_t)NASSIGN * 4;
  off = (off + 255) & ~(size_t)255;
  float* H = (float*)(ws + off);    off += (size_t)NASSIGN * I_DIM * 4;  // 23 MB
  float* Hs = (float*)(ws + off);   off += (size_t)T_TOK * SHI_DIM * 4;  // 11.5 MB
  float* R = (float*)(ws + off);    off += (size_t)NASSIGN * D_DIM * 4;  // 33.5 MB

  (void)hipMemsetAsync(counts, 0, E_EXP * sizeof(int), stream);

  gate_route_kernel<<<T_TOK, 32, 0, stream>>>(x, gate_w, gate_b, counts,
                                              tok_list, wt_list, tk_e, tk_s);
  scan_kernel<<<1, 32, 0, stream>>>(counts, offsets);

  shared_up_kernel<<<dim3(SHI_DIM / 64, T_TOK / 32), 256, 0, stream>>>(
      x, sw_gate, sw_up, Hs);
  shared_down_kernel<<<dim3(D_DIM / 64, T_TOK / 32), 256, 0, stream>>>(
      Hs, sw_down, out);

  moe_up_kernel<<<dim3(I_DIM / 64, T_TOK / 32, E_EXP), 256, 0, stream>>>(
      x, w_gate, w_up, counts, offsets, tok_list, H);
  moe_down_kernel<<<dim3(D_DIM / 64, T_TOK / 32, E_EXP), 256, 0, stream>>>(
      H, w_down, counts, offsets, wt_list, R);

  combine_kernel<<<T_TOK, 256, 0, stream>>>(offsets, tk_e, tk_s, R, out);
}